// MPNN_GAT_24850680775471
// MI455X (gfx1250) — compile-verified
//
#include <hip/hip_runtime.h>

// ---------------------------------------------------------------------------
// GAT (3 layers, fully-connected graph) for MI455X / gfx1250, wave32.
// Matrix work on V_WMMA_F32_16X16X4_F32 (fp32 in/out, matches reference dtype).
//
// WMMA f32 16x16x4 fragment layouts (CDNA5 ISA 7.12.2):
//   A (16x4, MxK):  lane l -> m = l%16 ; v0 = A[m][2*(l/16)], v1 = A[m][2*(l/16)+1]
//   B (4x16, KxN):  lane l -> n = l%16 ; v0 = B[2*(l/16)][n], v1 = B[2*(l/16)+1][n]
//   C/D (16x16):    vgpr r -> m = r + 8*(l/16), n = l%16
// ---------------------------------------------------------------------------

typedef float v2f __attribute__((ext_vector_type(2)));
typedef float v8f __attribute__((ext_vector_type(8)));

#define NHEADS 4
#define HID    64
#define NL     3
#define BATCH  64
#define NNODES 512
#define DIM    64
#define OUTD   (NHEADS * HID)   // 256
#define NEG_SLOPE 0.2f

__device__ __forceinline__ v8f wmma_f32(v2f a, v2f b, v8f c) {
  // 8 args: (neg_a, A, neg_b, B, c_mod, C, reuse_a, reuse_b)
  return __builtin_amdgcn_wmma_f32_16x16x4_f32(false, a, false, b, (short)0, c,
                                               false, false);
}

__device__ __forceinline__ float lrelu(float x) {
  return x >= 0.f ? x : NEG_SLOPE * x;
}

// ---------------------------------------------------------------------------
// h[row, o] = sum_k x[row, k] * W[o, k]     (row = b*N+n flattened, o in 0..255)
// Wave tile: 64 M x 32 O, K=64.  grid.x = (B*N)/64 blocks of 8 waves.
// ---------------------------------------------------------------------------
__global__ __launch_bounds__(256) void k_lin(const float* __restrict__ x,
                                             const float* __restrict__ W,
                                             float* __restrict__ hout) {
  const int tid  = threadIdx.x;
  const int lane = tid & 31;
  const int wid  = tid >> 5;
  const int half = lane >> 4;
  const int mr   = lane & 15;

  const int mtile = blockIdx.x;   // 64-row tile
  const int ctile = wid;          // 32-col tile (8 waves)
  const int row0  = mtile * 64;
  const int o0    = ctile * 32;

  const v8f vz = {0.f,0.f,0.f,0.f,0.f,0.f,0.f,0.f};
  v8f acc[4][2];
#pragma unroll
  for (int sm = 0; sm < 4; ++sm)
#pragma unroll
    for (int sn = 0; sn < 2; ++sn) acc[sm][sn] = vz;

  for (int kk = 0; kk < DIM; kk += 4) {
    const int kbase = kk + 2 * half;
    v2f afr[4], bfr[2];
#pragma unroll
    for (int sm = 0; sm < 4; ++sm)
      afr[sm] = *(const v2f*)(x + (size_t)(row0 + sm * 16 + mr) * DIM + kbase);
#pragma unroll
    for (int sn = 0; sn < 2; ++sn)
      bfr[sn] = *(const v2f*)(W + (size_t)(o0 + sn * 16 + mr) * DIM + kbase);
#pragma unroll
    for (int sm = 0; sm < 4; ++sm)
#pragma unroll
      for (int sn = 0; sn < 2; ++sn)
        acc[sm][sn] = wmma_f32(afr[sm], bfr[sn], acc[sm][sn]);
  }

#pragma unroll
  for (int sm = 0; sm < 4; ++sm)
#pragma unroll
    for (int sn = 0; sn < 2; ++sn)
#pragma unroll
      for (int r = 0; r < 8; ++r) {
        const int row = row0 + sm * 16 + r + 8 * half;
        const int col = o0 + sn * 16 + mr;
        hout[(size_t)row * OUTD + col] = acc[sm][sn][r];
      }
}

// ---------------------------------------------------------------------------
// a_src[b,h,n] = <h[b,n,h,:], att_src[h,:]> ; same for a_dst.  Wave per node.
// ---------------------------------------------------------------------------
__global__ __launch_bounds__(256) void k_scores(const float* __restrict__ hbuf,
                                                const float* __restrict__ attS,
                                                const float* __restrict__ attD,
                                                float* __restrict__ asrc,
                                                float* __restrict__ adst) {
  const int tid  = threadIdx.x;
  const int lane = tid & 31;
  const int wid  = tid >> 5;
  const int node = blockIdx.x * 8 + wid;      // 0 .. B*N-1
  const int b = node >> 9, n = node & 511;
  const float* hrow = hbuf + (size_t)node * OUTD;

  for (int hh = 0; hh < NHEADS; ++hh) {
    const float h0 = hrow[hh * HID + lane];
    const float h1 = hrow[hh * HID + 32 + lane];
    float ss = h0 * attS[hh * HID + lane] + h1 * attS[hh * HID + 32 + lane];
    float sd = h0 * attD[hh * HID + lane] + h1 * attD[hh * HID + 32 + lane];
#pragma unroll
    for (int off = 16; off; off >>= 1) {
      ss += __shfl_xor(ss, off, 32);
      sd += __shfl_xor(sd, off, 32);
    }
    if (lane == 0) {
      asrc[((b * NHEADS + hh) << 9) + n] = ss;
      adst[((b * NHEADS + hh) << 9) + n] = sd;
    }
  }
}

// ---------------------------------------------------------------------------
// Attention softmax + aggregation + head-mean + bias + ReLU, fused.
// Block: (b, 128-row i tile); 8 waves x 16 rows each; loops 4 heads.
// P[i][j] = exp(lrelu(d_i + s_j) - lrelu(d_i + max_j s_j))  (LeakyReLU monotone
// => analytic row max).  out_h = P @ H via WMMA; Z from summing P fragments.
// LDS: H chunk 128x64 with stride 80 -> bank (16r+c)%64, conflict-free halves.
// ---------------------------------------------------------------------------
__global__ __launch_bounds__(256) void k_agg(const float* __restrict__ hbuf,
                                             const float* __restrict__ asrc,
                                             const float* __restrict__ adst,
                                             const float* __restrict__ bias,
                                             float* __restrict__ xout) {
  __shared__ float sH[128 * 80];
  __shared__ float sA[512];
  __shared__ float sred[8];

  const int tid  = threadIdx.x;
  const int lane = tid & 31;
  const int wid  = tid >> 5;
  const int half = lane >> 4;
  const int mr   = lane & 15;

  const int b    = blockIdx.y;
  const int i0   = blockIdx.x * 128;
  const int mrow = i0 + wid * 16 + mr;   // this lane's A-fragment row

  const v8f vz = {0.f,0.f,0.f,0.f,0.f,0.f,0.f,0.f};
  v8f accm[4];
#pragma unroll
  for (int t = 0; t < 4; ++t) accm[t] = vz;

  for (int hh = 0; hh < NHEADS; ++hh) {
    const int bh = (b * NHEADS + hh) << 9;

    __syncthreads();                     // previous head done with sA/sred
    sA[tid]       = asrc[bh + tid];
    sA[tid + 256] = asrc[bh + tid + 256];
    __syncthreads();

    // block max of a_src row
    float mx = fmaxf(sA[tid], sA[tid + 256]);
#pragma unroll
    for (int off = 16; off; off >>= 1) mx = fmaxf(mx, __shfl_xor(mx, off, 32));
    if (lane == 0) sred[wid] = mx;
    __syncthreads();
    float smax = sred[0];
#pragma unroll
    for (int w = 1; w < 8; ++w) smax = fmaxf(smax, sred[w]);

    const float d     = adst[bh + mrow];
    const float mterm = lrelu(d + smax);  // = max_j lrelu(d + s_j)

    v8f acc[4];
#pragma unroll
    for (int t = 0; t < 4; ++t) acc[t] = vz;
    float zacc = 0.f;

    for (int jc = 0; jc < 4; ++jc) {
      const int j0 = jc * 128;
      __syncthreads();                   // prior chunk reads done
#pragma unroll
      for (int k = 0; k < 32; ++k) {     // stage H[b, j0:j0+128, hh*64:+64]
        const int e  = tid + k * 256;
        const int rr = e >> 6, cc = e & 63;
        sH[rr * 80 + cc] =
            hbuf[(size_t)(b * NNODES + j0 + rr) * OUTD + hh * HID + cc];
      }
      __syncthreads();

      for (int jj = 0; jj < 128; jj += 4) {
        const int ja = j0 + jj + 2 * half;
        const float s0 = sA[ja];
        const float s1 = sA[ja + 1];
        const float p0 = __expf(lrelu(d + s0) - mterm);
        const float p1 = __expf(lrelu(d + s1) - mterm);
        zacc += p0 + p1;
        v2f af; af.x = p0; af.y = p1;
        const int r0 = (jj + 2 * half) * 80 + mr;
#pragma unroll
        for (int t = 0; t < 4; ++t) {
          v2f bf;
          bf.x = sH[r0 + t * 16];
          bf.y = sH[r0 + 80 + t * 16];
          acc[t] = wmma_f32(af, bf, acc[t]);
        }
      }
    }

    // Z[m] lives in lane m (and m+16): combine halves, then broadcast per vgpr row
    const float z = zacc + __shfl_xor(zacc, 16, 32);
#pragma unroll
    for (int r = 0; r < 8; ++r) {
      const float zr = __shfl(z, r + 8 * half, 32);
      const float sc = 0.25f / zr;       // head-mean (1/NHEADS) folded in
#pragma unroll
      for (int t = 0; t < 4; ++t) accm[t][r] += acc[t][r] * sc;
    }
  }

  // bias + ReLU + store x_next[b, i, c]
#pragma unroll
  for (int t = 0; t < 4; ++t) {
    const float bsv = bias[t * 16 + mr];
#pragma unroll
    for (int r = 0; r < 8; ++r) {
      float v = accm[t][r] + bsv;
      v = v > 0.f ? v : 0.f;
      xout[(size_t)(b * NNODES + i0 + wid * 16 + r + 8 * half) * HID +
           t * 16 + mr] = v;
    }
  }
}

// ---------------------------------------------------------------------------
// pooled[b,:] = mean_i x[b,i,:]; out = pooled @ Wr^T + br.  One block per b.
// ---------------------------------------------------------------------------
__global__ __launch_bounds__(64) void k_readout(const float* __restrict__ x,
                                                const float* __restrict__ Wr,
                                                const float* __restrict__ br,
                                                float* __restrict__ out) {
  __shared__ float pooled[DIM];
  const int b = blockIdx.x, d = threadIdx.x;
  float s = 0.f;
  for (int i = 0; i < NNODES; ++i)
    s += x[(size_t)(b * NNODES + i) * DIM + d];
  pooled[d] = s * (1.f / (float)NNODES);
  __syncthreads();
  float o = br[d];
  for (int k = 0; k < DIM; ++k) o += pooled[k] * Wr[d * DIM + k];
  out[b * DIM + d] = o;
}

// ---------------------------------------------------------------------------
extern "C" void kernel_launch(void* const* d_in, const int* in_sizes, int n_in,
                              void* d_out, int out_size, void* d_ws,
                              size_t ws_size, hipStream_t stream) {
  (void)in_sizes; (void)n_in; (void)out_size; (void)ws_size;
  const float* emb    = (const float*)d_in[0];   // [B,N,64]
  const float* lin_w  = (const float*)d_in[1];   // [3,256,64]
  const float* attS   = (const float*)d_in[2];   // [3,4,64]
  const float* attD   = (const float*)d_in[3];   // [3,4,64]
  const float* conv_b = (const float*)d_in[4];   // [3,64]
  const float* rw     = (const float*)d_in[5];   // [64,64]
  const float* rb     = (const float*)d_in[6];   // [64]

  float* ws = (float*)d_ws;
  const size_t XSZ = (size_t)BATCH * NNODES * DIM;    // 2,097,152
  const size_t HSZ = (size_t)BATCH * NNODES * OUTD;   // 8,388,608
  const size_t ASZ = (size_t)BATCH * NHEADS * NNODES; // 131,072
  float* xbuf0 = ws;
  float* xbuf1 = ws + XSZ;
  float* hbuf  = ws + 2 * XSZ;
  float* asrc  = hbuf + HSZ;
  float* adst  = asrc + ASZ;

  const float* xin = emb;
  float* xo[NL] = {xbuf0, xbuf1, xbuf0};

  for (int l = 0; l < NL; ++l) {
    k_lin<<<(BATCH * NNODES) / 64, 256, 0, stream>>>(
        xin, lin_w + (size_t)l * OUTD * DIM, hbuf);
    k_scores<<<(BATCH * NNODES) / 8, 256, 0, stream>>>(
        hbuf, attS + l * NHEADS * HID, attD + l * NHEADS * HID, asrc, adst);
    k_agg<<<dim3(NNODES / 128, BATCH), 256, 0, stream>>>(
        hbuf, asrc, adst, conv_b + l * HID, xo[l]);
    xin = xo[l];
  }
  k_readout<<<BATCH, 64, 0, stream>>>(xin, rw, rb, (float*)d_out);
}